// Attention_24275155157367
// MI455X (gfx1250) — compile-verified
//
#include <hip/hip_runtime.h>
#include <math.h>

typedef __attribute__((ext_vector_type(2))) float v2f;
typedef __attribute__((ext_vector_type(8))) float v8f;

__device__ __forceinline__ v8f wmma4(v2f a, v2f b, v8f c) {
  // D = A(16x4,f32) * B(4x16,f32) + C(16x16,f32)
  return __builtin_amdgcn_wmma_f32_16x16x4_f32(false, a, false, b, (short)0, c, false, false);
}

#define NTOK   577
#define NPAD   592
#define NTILES 37
#define ROWS   18464   // 32*577

// ---------------------------------------------------------------------------
// Fold Wqk with rep_k / rep_q (and SCALE + biases); transpose Wv.
// wq/wk: [6][64][16] (cols 8..15 zero), bq/bk: [6][16], wvT: [6][64][64]
// ---------------------------------------------------------------------------
__global__ void prep_kernel(const float* __restrict__ Wqk, const float* __restrict__ bqk,
                            const float* __restrict__ Wv,  const float* __restrict__ rep_k,
                            const float* __restrict__ rep_q,
                            float* wq, float* wk, float* bq, float* bk, float* wvT) {
  const float SCALE = 0.05103103630798287f; // 384^-0.5
  int tid = threadIdx.x;
  for (int i = tid; i < 6144; i += 256) {
    int h = i >> 10, rem = i & 1023, cc = rem >> 4, r = rem & 15;
    float sq = 0.f, sk = 0.f;
    if (r < 8) {
      int qb = (h >> 1) * 8192 + (h & 1) * 4096;        // Wqk[h/2][(h%2)*64 + co][..]
      int kb = (3 + (h >> 1)) * 8192 + (h & 1) * 4096;
      for (int co = 0; co < 64; ++co) {
        sq += Wqk[qb + co * 64 + cc] * rep_k[h * 512 + co * 8 + r];
        sk += rep_q[h * 512 + r * 64 + co] * Wqk[kb + co * 64 + cc];
      }
      sq *= SCALE; sk *= SCALE;
    }
    wq[i] = sq; wk[i] = sk;
  }
  for (int i = tid; i < 96; i += 256) {
    int h = i >> 4, r = i & 15;
    float sq = 0.f, sk = 0.f;
    if (r < 8) {
      for (int co = 0; co < 64; ++co) {
        sq += bqk[(h >> 1) * 128 + (h & 1) * 64 + co] * rep_k[h * 512 + co * 8 + r];
        sk += rep_q[h * 512 + r * 64 + co] * bqk[(3 + (h >> 1)) * 128 + (h & 1) * 64 + co];
      }
      sq *= SCALE; sk *= SCALE;
    }
    bq[i] = sq; bk[i] = sk;
  }
  for (int i = tid; i < 24576; i += 256) {
    int h = i >> 12, rem = i & 4095, cc = rem >> 6, c = rem & 63;
    wvT[i] = Wv[h * 4096 + c * 64 + cc];  // wvT[h][cc][c]
  }
}

// ---------------------------------------------------------------------------
// LayerNorm over last dim (384). One block per row.
// ---------------------------------------------------------------------------
__global__ void ln_kernel(const float* __restrict__ x, const float* __restrict__ g,
                          const float* __restrict__ bvec, float* __restrict__ y) {
  __shared__ float s1[512];
  __shared__ float s2[512];
  int tid = threadIdx.x;
  size_t row = blockIdx.x;
  float v = 0.f;
  if (tid < 384) v = x[row * 384 + tid];
  s1[tid] = v; s2[tid] = v * v;
  __syncthreads();
  for (int s = 256; s > 0; s >>= 1) {
    if (tid < s) { s1[tid] += s1[tid + s]; s2[tid] += s2[tid + s]; }
    __syncthreads();
  }
  float mean = s1[0] * (1.0f / 384.0f);
  float var  = s2[0] * (1.0f / 384.0f) - mean * mean;
  float rs = rsqrtf(var + 1e-5f);
  if (tid < 384) y[row * 384 + tid] = (v - mean) * rs * g[tid] + bvec[tid];
}

// ---------------------------------------------------------------------------
// Projections: ql[b][h][n][8], klT[b][h][n][8], vA[b][h][n][64] (n padded to 592)
// Block = (16-row tile of n) x b; 4 waves share 36 WMMA output tiles.
// ---------------------------------------------------------------------------
__global__ void proj_kernel(const float* __restrict__ xn,
                            const float* __restrict__ wq, const float* __restrict__ wk,
                            const float* __restrict__ bq, const float* __restrict__ bk,
                            const float* __restrict__ wvT, const float* __restrict__ bv,
                            float* __restrict__ ql, float* __restrict__ kl,
                            float* __restrict__ vA) {
  __shared__ float As[16 * 388];
  int tid = threadIdx.x;
  int nb = blockIdx.x * 16;
  int b  = blockIdx.y;
  for (int i = tid; i < 6144; i += 128) {
    int r = i / 384, k = i - r * 384;
    long gr = (long)b * NTOK + nb + r;
    As[r * 388 + k] = (gr < ROWS) ? xn[(size_t)gr * 384 + k] : 0.f;
  }
  __syncthreads();
  int l = tid & 31, wv = tid >> 5;
  int lm = l & 15, koff = (l >> 4) * 2;
  for (int job = wv; job < 36; job += 4) {
    v8f acc = {};
    if (job < 12) {                       // ql (job<6) / klT
      bool isq = job < 6;
      int h = isq ? job : job - 6;
      int base = (isq ? 0 : 192) + (h >> 1) * 64;
      const float* W = isq ? wq : wk;
      for (int ks = 0; ks < 16; ++ks) {
        int kk = ks * 4 + koff;
        v2f a = *(const v2f*)&As[lm * 388 + base + kk];
        v2f bb; bb.x = W[h * 1024 + kk * 16 + lm];
                bb.y = W[h * 1024 + (kk + 1) * 16 + lm];
        acc = wmma4(a, bb, acc);
      }
      if (lm < 8) {
        float bias = (isq ? bq : bk)[h * 16 + lm];
        float* out = isq ? ql : kl;
        size_t rb = (size_t)(b * 6 + h) * NPAD + nb;
        for (int j = 0; j < 8; ++j) {
          int row = j + 8 * (l >> 4);
          out[(rb + row) * 8 + lm] = acc[j] + bias;
        }
      }
    } else {                               // V with channel permutation d = h + 6*c
      int vj = job - 12;
      int h = vj >> 2, ct = vj & 3;
      int base = h * 64;
      for (int ks = 0; ks < 16; ++ks) {
        int kk = ks * 4 + koff;
        v2f a = *(const v2f*)&As[lm * 388 + base + kk];
        v2f bb; bb.x = wvT[h * 4096 + kk * 64 + ct * 16 + lm];
                bb.y = wvT[h * 4096 + (kk + 1) * 64 + ct * 16 + lm];
        acc = wmma4(a, bb, acc);
      }
      int c = ct * 16 + lm;
      int d = h + 6 * c;
      int hh = d >> 6, c2 = d & 63;
      float bias = bv[h * 64 + c];
      size_t ob = (size_t)(b * 6 + hh) * NPAD + nb;
      for (int j = 0; j < 8; ++j) {
        int row = j + 8 * (l >> 4);
        vA[(ob + row) * 64 + c2] = acc[j] + bias;
      }
    }
  }
}

// ---------------------------------------------------------------------------
// Attention: block = (16-query tile, head, batch). Logits via K=8 WMMA,
// softmax in LDS, P@V via K=592 WMMA. Writes o in final (B,N,384) order.
// ---------------------------------------------------------------------------
__global__ void attn_kernel(const float* __restrict__ ql, const float* __restrict__ kl,
                            const float* __restrict__ vA, float* __restrict__ o) {
  __shared__ float S[16 * 596];
  __shared__ float red[16][8];
  __shared__ float rstat[16];
  int tid = threadIdx.x;
  int b = blockIdx.z, h = blockIdx.y;
  int nb = blockIdx.x * 16;
  int l = tid & 31, wv = tid >> 5, lm = l & 15, koff = (l >> 4) * 2;
  size_t hb = (size_t)(b * 6 + h) * NPAD;
  const float* qlp = ql + hb * 8;
  const float* klp = kl + hb * 8;
  const float* vp  = vA + hb * 64;
  int qrow = nb + lm;
  v2f aq01 = *(const v2f*)&qlp[qrow * 8 + koff];
  v2f aq23 = *(const v2f*)&qlp[qrow * 8 + 4 + koff];
  for (int mt = wv; mt < NTILES; mt += 4) {
    int mb = mt * 16, mrow = mb + lm;
    v8f acc = {};
    v2f b0 = *(const v2f*)&klp[mrow * 8 + koff];
    v2f b1 = *(const v2f*)&klp[mrow * 8 + 4 + koff];
    acc = wmma4(aq01, b0, acc);
    acc = wmma4(aq23, b1, acc);
    for (int j = 0; j < 8; ++j)
      S[(j + 8 * (l >> 4)) * 596 + mb + lm] = acc[j];
  }
  __syncthreads();
  // softmax over m (mask m>=577), 8 threads per row
  int r = tid >> 3, s8 = tid & 7;
  float mx = -3.4e38f;
  for (int m = s8; m < NTOK; m += 8) mx = fmaxf(mx, S[r * 596 + m]);
  red[r][s8] = mx;
  __syncthreads();
  if (s8 == 0) { float t = red[r][0]; for (int i = 1; i < 8; ++i) t = fmaxf(t, red[r][i]); rstat[r] = t; }
  __syncthreads();
  float mr = rstat[r], sum = 0.f;
  for (int m = s8; m < NTOK; m += 8) { float p = __expf(S[r * 596 + m] - mr); S[r * 596 + m] = p; sum += p; }
  for (int m = NTOK + s8; m < NPAD; m += 8) S[r * 596 + m] = 0.f;
  __syncthreads();
  red[r][s8] = sum;
  __syncthreads();
  if (s8 == 0) { float t = 0.f; for (int i = 0; i < 8; ++i) t += red[r][i]; rstat[r] = t; }
  __syncthreads();
  float inv = 1.0f / rstat[r];
  for (int m = s8; m < NPAD; m += 8) S[r * 596 + m] *= inv;
  __syncthreads();
  // O(16x64) = P(16x592) @ V(592x64); wave w owns 16-column tile w
  int c2 = wv * 16 + lm;
  v8f acc = {};
  for (int ks = 0; ks < NPAD / 4; ++ks) {
    int kk = ks * 4 + koff;
    v2f a = *(const v2f*)&S[lm * 596 + kk];
    v2f bb; bb.x = vp[(size_t)kk * 64 + c2];
            bb.y = vp[(size_t)(kk + 1) * 64 + c2];
    acc = wmma4(a, bb, acc);
  }
  for (int j = 0; j < 8; ++j) {
    int row = nb + j + 8 * (l >> 4);
    if (row < NTOK)
      o[((size_t)b * NTOK + row) * 384 + h * 64 + c2] = acc[j];
  }
}

// ---------------------------------------------------------------------------
// 18464x384x384 GEMM, 32 rows x 64 cols per block (4 waves, 2 row-tiles/wave
// sharing each B fragment -> half the W traffic, 2 independent WMMA chains).
// mode 0: out = gelu(A@W + bias); mode 1: out = A@W + bias + resid
// ---------------------------------------------------------------------------
__global__ void gemm_kernel(const float* __restrict__ A, const float* __restrict__ W,
                            const float* __restrict__ bias, const float* __restrict__ resid,
                            float* __restrict__ out, int mode) {
  __shared__ float As[32 * 388];
  int tid = threadIdx.x;
  size_t rb = (size_t)blockIdx.x * 32;
  for (int i = tid; i < 12288; i += 128) {
    int r = i / 384, k = i - r * 384;
    As[r * 388 + k] = A[(rb + r) * 384 + k];
  }
  __syncthreads();
  int l = tid & 31, wv = tid >> 5;
  int lm = l & 15, koff = (l >> 4) * 2;
  int col = blockIdx.y * 64 + wv * 16 + lm;
  v8f acc0 = {}, acc1 = {};
  for (int ks = 0; ks < 96; ++ks) {
    int kk = ks * 4 + koff;
    v2f a0 = *(const v2f*)&As[lm * 388 + kk];
    v2f a1 = *(const v2f*)&As[(lm + 16) * 388 + kk];
    v2f bb; bb.x = W[(size_t)kk * 384 + col];
            bb.y = W[(size_t)(kk + 1) * 384 + col];
    acc0 = wmma4(a0, bb, acc0);
    acc1 = wmma4(a1, bb, acc1);
  }
  float bs = bias[col];
  for (int j = 0; j < 8; ++j) {
    size_t row0 = rb + j + 8 * (l >> 4);
    size_t row1 = row0 + 16;
    float v0 = acc0[j] + bs;
    float v1 = acc1[j] + bs;
    if (mode == 0) {
      v0 = 0.5f * v0 * (1.0f + erff(v0 * 0.70710678118654752f));
      v1 = 0.5f * v1 * (1.0f + erff(v1 * 0.70710678118654752f));
    } else {
      v0 += resid[row0 * 384 + col];
      v1 += resid[row1 * 384 + col];
    }
    out[row0 * 384 + col] = v0;
    out[row1 * 384 + col] = v1;
  }
}

// ---------------------------------------------------------------------------
extern "C" void kernel_launch(void* const* d_in, const int* in_sizes, int n_in,
                              void* d_out, int out_size, void* d_ws, size_t ws_size,
                              hipStream_t stream) {
  const float* x     = (const float*)d_in[0];
  const float* ln1_g = (const float*)d_in[1];
  const float* ln1_b = (const float*)d_in[2];
  const float* Wqk   = (const float*)d_in[3];
  const float* bqk   = (const float*)d_in[4];
  const float* Wv    = (const float*)d_in[5];
  const float* bv    = (const float*)d_in[6];
  const float* rep_k = (const float*)d_in[7];
  const float* rep_q = (const float*)d_in[8];
  const float* ln2_g = (const float*)d_in[9];
  const float* ln2_b = (const float*)d_in[10];
  const float* W1    = (const float*)d_in[11];
  const float* b1    = (const float*)d_in[12];
  const float* W2    = (const float*)d_in[13];
  const float* b2    = (const float*)d_in[14];
  float* out = (float*)d_out;
  float* w = (float*)d_ws;

  // workspace layout (floats); total ~23.31M floats (~93.3 MB)
  float* xn   = w + 0;          // 7,090,176 (B,N,384); reused as o_pre after attention
  float* ql   = w + 7090176;    //   909,312 (B,6,592,8)
  float* kl   = w + 7999488;    //   909,312 (B,6,592,8)
  float* vA   = w + 8908800;    // 7,274,496 (B,6,592,64); reused as h1 after attention
  float* ln2o = w + 16183296;   // 7,090,176
  float* wq   = w + 23273472;   // 6,144
  float* wk   = w + 23279616;   // 6,144
  float* bq   = w + 23285760;   // 96
  float* bk   = w + 23285856;   // 96
  float* wvT  = w + 23285952;   // 24,576
  float* h1   = vA;

  prep_kernel<<<1, 256, 0, stream>>>(Wqk, bqk, Wv, rep_k, rep_q, wq, wk, bq, bk, wvT);
  ln_kernel<<<ROWS, 512, 0, stream>>>(x, ln1_g, ln1_b, xn);
  proj_kernel<<<dim3(NTILES, 32), 128, 0, stream>>>(xn, wq, wk, bq, bk, wvT, bv, ql, kl, vA);
  attn_kernel<<<dim3(NTILES, 6, 32), 128, 0, stream>>>(ql, kl, vA, xn /* o_pre */);
  ln_kernel<<<ROWS, 512, 0, stream>>>(xn, ln2_g, ln2_b, ln2o);
  gemm_kernel<<<dim3(ROWS / 32, 6), 128, 0, stream>>>(ln2o, W1, b1, nullptr, h1, 0);
  gemm_kernel<<<dim3(ROWS / 32, 6), 128, 0, stream>>>(h1, W2, b2, x, out, 1);
}